// RGAT_20401094656590
// MI455X (gfx1250) — compile-verified
//
#include <hip/hip_runtime.h>

// ---------------- problem constants ----------------
#define NN    30000
#define EE    200000
#define RR    2
#define LL    2
#define FDIM  256     // IN == HID
#define HH    4
#define DHH   64
#define CC    40
#define SLOPE 0.2f

typedef float v2f __attribute__((ext_vector_type(2)));
typedef float v8f __attribute__((ext_vector_type(8)));

// ---------------- gfx1250 async memory->LDS staging (probe via __has_builtin) ---
#ifndef __has_builtin
#define __has_builtin(x) 0
#endif
#if __has_builtin(__builtin_amdgcn_global_load_async_to_lds_b128)
#define HAVE_ASYNC_LDS 1
typedef int v4i __attribute__((vector_size(16)));       // matches builtin's pointee
typedef __attribute__((address_space(1))) v4i glb_v4i;  // global (AS1) int4
typedef __attribute__((address_space(3))) v4i lds_v4i;  // LDS (AS3) int4
#else
#define HAVE_ASYNC_LDS 0
#endif

__device__ __forceinline__ void async_wait0() {
#if __has_builtin(__builtin_amdgcn_s_wait_asynccnt)
  __builtin_amdgcn_s_wait_asynccnt(0);
#else
  asm volatile("s_wait_asynccnt 0x0" ::: "memory");
#endif
}

// ---------------- WMMA fp32 GEMM:  C[nrows,256] = A[nrows,256] @ B[256,256] -----
// Block: 256 threads = 8 waves. Block tile: 128 rows x 64 cols (grid.y = 4).
// A staged through LDS in K-chunks of 64 (stride padded to 68 floats -> 16
// distinct banks for the 16-row fragment reads).
#define KC   64
#define LDSP (KC + 4)

// Stage A[rowBase:+128, kc:+64] into lds (128 x LDSP floats). Uses the CDNA5
// async memory->LDS path when the toolchain exposes it (ASYNCcnt-tracked,
// no VGPR round-trip); falls back to b128 load + ds_store otherwise.
__device__ __forceinline__ void stage_tile(const float* __restrict__ A, float* lds,
                                           int rowBase, int nrows, int kc, int tid) {
#if HAVE_ASYNC_LDS
  // Escape the LDS array so alias analysis knows the async engine writes it
  // (the builtin's AS3 pointer is fabricated from a raw offset, otherwise the
  // array looks never-stored and its loads would fold away / hoist).
  asm volatile("" : "+v"(lds) :: "memory");
#endif
#pragma unroll
  for (int i = 0; i < 8; ++i) {
    int idx = i * 256 + tid;           // float4 index within 128x16 tile
    int r = idx >> 4, c4 = idx & 15;
    int gr = rowBase + r;
#if HAVE_ASYNC_LDS
    int grc = gr < nrows ? gr : (nrows - 1);  // clamp: tail rows never consumed
    unsigned ldsoff = (unsigned)(r * LDSP + c4 * 4) * 4u;  // single LDS array @ base 0
    __builtin_amdgcn_global_load_async_to_lds_b128(
        (glb_v4i*)(uintptr_t)(A + (size_t)grc * FDIM + kc + c4 * 4),
        (lds_v4i*)(uintptr_t)ldsoff, 0, 0);
#else
    float4 v = make_float4(0.f, 0.f, 0.f, 0.f);
    if (gr < nrows) v = *(const float4*)(A + (size_t)gr * FDIM + kc + c4 * 4);
    *(float4*)(&lds[r * LDSP + c4 * 4]) = v;
#endif
  }
#if HAVE_ASYNC_LDS
  async_wait0();
  asm volatile("" ::: "memory");
#endif
  __syncthreads();
}

__global__ __launch_bounds__(256) void k_gemm256(const float* __restrict__ A,
                                                 const float* __restrict__ B,
                                                 float* __restrict__ C, int nrows) {
  __shared__ float lds[128 * LDSP];
  const int tid = threadIdx.x;
  const int rowBase = blockIdx.x * 128;
  const int colBase = blockIdx.y * 64;
  const int wave = tid >> 5, lane = tid & 31;
  const int l15 = lane & 15, lhalf = lane >> 4;

  v8f acc[4] = {};

  for (int kc = 0; kc < FDIM; kc += KC) {
    stage_tile(A, lds, rowBase, nrows, kc, tid);

    // A frag (ISA 16x4 layout): lanes 0-15 carry K=kk+0/kk+1, lanes 16-31 K=kk+2/kk+3
    const float* ap  = &lds[(wave * 16 + l15) * LDSP + 2 * lhalf];
    const float* bp0 = B + (size_t)(kc + 2 * lhalf) * FDIM + colBase + l15;
    for (int k = 0; k < KC; k += 4) {
      float2 av = *(const float2*)(ap + k);   // ds_load_b64
      v2f a; a.x = av.x; a.y = av.y;
      const float* bp = bp0 + (size_t)k * FDIM;
#pragma unroll
      for (int t = 0; t < 4; ++t) {
        v2f b; b.x = bp[t * 16]; b.y = bp[FDIM + t * 16];
        acc[t] = __builtin_amdgcn_wmma_f32_16x16x4_f32(
            false, a, false, b, (short)0, acc[t], false, false);
      }
    }
    __syncthreads();
  }

  // C/D layout: VGPR j -> M=j (lanes 0-15) / M=j+8 (lanes 16-31); N = l15
  const int orow = rowBase + wave * 16 + 8 * lhalf;
#pragma unroll
  for (int j = 0; j < 8; ++j) {
    int r = orow + j;
    if (r < nrows) {
      float* cp = C + (size_t)r * FDIM + colBase + l15;
      cp[0]  = acc[0][j];
      cp[16] = acc[1][j];
      cp[32] = acc[2][j];
      cp[48] = acc[3][j];
    }
  }
}

// ---------------- classifier GEMM:  out[nrows,40] = A[nrows,256] @ Wc[256,40] + bc
__global__ __launch_bounds__(256) void k_gemm_cls(const float* __restrict__ A,
                                                  const float* __restrict__ Wc,
                                                  const float* __restrict__ bc,
                                                  float* __restrict__ Out, int nrows) {
  __shared__ float lds[128 * LDSP];
  const int tid = threadIdx.x;
  const int rowBase = blockIdx.x * 128;
  const int wave = tid >> 5, lane = tid & 31;
  const int l15 = lane & 15, lhalf = lane >> 4;

  v8f acc[3] = {};   // 3 col-tiles cover 48 >= 40 cols

  for (int kc = 0; kc < FDIM; kc += KC) {
    stage_tile(A, lds, rowBase, nrows, kc, tid);

    const float* ap = &lds[(wave * 16 + l15) * LDSP + 2 * lhalf];
    for (int k = 0; k < KC; k += 4) {
      float2 av = *(const float2*)(ap + k);
      v2f a; a.x = av.x; a.y = av.y;
      int krow = kc + k + 2 * lhalf;
#pragma unroll
      for (int t = 0; t < 3; ++t) {
        int col = t * 16 + l15;
        v2f b;
        b.x = (col < CC) ? Wc[(size_t)krow * CC + col] : 0.f;
        b.y = (col < CC) ? Wc[(size_t)(krow + 1) * CC + col] : 0.f;
        acc[t] = __builtin_amdgcn_wmma_f32_16x16x4_f32(
            false, a, false, b, (short)0, acc[t], false, false);
      }
    }
    __syncthreads();
  }

  const int orow = rowBase + wave * 16 + 8 * lhalf;
#pragma unroll
  for (int j = 0; j < 8; ++j) {
    int r = orow + j;
    if (r < nrows) {
#pragma unroll
      for (int t = 0; t < 3; ++t) {
        int col = t * 16 + l15;
        if (col < CC) Out[(size_t)r * CC + col] = acc[t][j] + bc[col];
      }
    }
  }
}

// ---------------- per-node attention scores:  el/er[n,h] = sum_dh hp[n,h,dh]*a[h,dh]
// one wave per node; lane covers 8 consecutive features (head = lane>>3)
__global__ __launch_bounds__(256) void k_attn_scores(const float* __restrict__ hp,
                                                     const float* __restrict__ al,
                                                     const float* __restrict__ ar,
                                                     float* __restrict__ el,
                                                     float* __restrict__ er, int n) {
  int wid = (blockIdx.x * blockDim.x + threadIdx.x) >> 5;
  int lane = threadIdx.x & 31;
  if (wid >= n) return;
  const float* h = hp + (size_t)wid * FDIM + lane * 8;
  float4 a0 = *(const float4*)h, a1 = *(const float4*)(h + 4);
  const float* alp = al + lane * 8;
  const float* arp = ar + lane * 8;
  float pl = a0.x * alp[0] + a0.y * alp[1] + a0.z * alp[2] + a0.w * alp[3] +
             a1.x * alp[4] + a1.y * alp[5] + a1.z * alp[6] + a1.w * alp[7];
  float pr = a0.x * arp[0] + a0.y * arp[1] + a0.z * arp[2] + a0.w * arp[3] +
             a1.x * arp[4] + a1.y * arp[5] + a1.z * arp[6] + a1.w * arp[7];
#pragma unroll
  for (int m = 1; m < 8; m <<= 1) {
    pl += __shfl_xor(pl, m);
    pr += __shfl_xor(pr, m);
  }
  if ((lane & 7) == 0) {
    int head = lane >> 3;
    el[wid * HH + head] = pl;
    er[wid * HH + head] = pr;
  }
}

// ---------------- order-preserving float<->int for lock-free float atomicMax
__device__ __forceinline__ int f2ord(float f) {
  int i = __float_as_int(f);
  return i >= 0 ? i : (i ^ 0x7fffffff);
}
__device__ __forceinline__ float ord2f(int i) {
  return __int_as_float(i >= 0 ? i : (i ^ 0x7fffffff));
}

__global__ void k_init_out(float* __restrict__ out, const float* __restrict__ b0,
                           const float* __restrict__ b1, int total) {
  int i = blockIdx.x * blockDim.x + threadIdx.x;
  if (i < total) out[i] = b0[i & (FDIM - 1)] + b1[i & (FDIM - 1)];
}

__global__ void k_init_nh(int* __restrict__ mi, float* __restrict__ den, int total) {
  int i = blockIdx.x * blockDim.x + threadIdx.x;
  if (i < total) { mi[i] = (int)0x80000000; den[i] = 0.f; }
}

// edge pass 1: e = leakyrelu(el[s]+er[d]); segment-max via int atomicMax
__global__ void k_edge_max(const int* __restrict__ src, const int* __restrict__ dst,
                           const float* __restrict__ el, const float* __restrict__ er,
                           float* __restrict__ ebuf, int* __restrict__ mi, int ne) {
  int e = blockIdx.x * blockDim.x + threadIdx.x;
  if (e >= ne) return;
  int s = src[e], d = dst[e];
  float4 ls = *(const float4*)(el + (size_t)s * HH);
  float4 rd = *(const float4*)(er + (size_t)d * HH);
  float v[4] = {ls.x + rd.x, ls.y + rd.y, ls.z + rd.z, ls.w + rd.w};
  float4 o;
#pragma unroll
  for (int h = 0; h < HH; ++h) {
    float x = v[h];
    x = x > 0.f ? x : SLOPE * x;
    v[h] = x;
    atomicMax(&mi[(size_t)d * HH + h], f2ord(x));
  }
  o.x = v[0]; o.y = v[1]; o.z = v[2]; o.w = v[3];
  *(float4*)(ebuf + (size_t)e * HH) = o;
}

// edge pass 2: ex = exp(e - m[d]); denom[d] += ex
__global__ void k_edge_exp(const int* __restrict__ dst, const int* __restrict__ mi,
                           float* __restrict__ ebuf, float* __restrict__ den, int ne) {
  int e = blockIdx.x * blockDim.x + threadIdx.x;
  if (e >= ne) return;
  int d = dst[e];
  float4 ev = *(const float4*)(ebuf + (size_t)e * HH);
  float v[4] = {ev.x, ev.y, ev.z, ev.w};
#pragma unroll
  for (int h = 0; h < HH; ++h) {
    float m = ord2f(mi[(size_t)d * HH + h]);
    float ex = __expf(v[h] - m);
    v[h] = ex;
    atomicAdd(&den[(size_t)d * HH + h], ex);
  }
  float4 o; o.x = v[0]; o.y = v[1]; o.z = v[2]; o.w = v[3];
  *(float4*)(ebuf + (size_t)e * HH) = o;
}

// edge pass 3: out[d] += (ex/denom[d]) * hp[s]   (one wave per edge, 8 f32/lane)
__global__ __launch_bounds__(256) void k_edge_aggr(const int* __restrict__ src,
                                                   const int* __restrict__ dst,
                                                   const float* __restrict__ ebuf,
                                                   const float* __restrict__ den,
                                                   const float* __restrict__ hp,
                                                   float* __restrict__ out, int ne) {
  int e = (blockIdx.x * blockDim.x + threadIdx.x) >> 5;
  int lane = threadIdx.x & 31;
  if (e >= ne) return;
  int s = src[e], d = dst[e];
  int head = lane >> 3;
  float alpha = ebuf[(size_t)e * HH + head] / (den[(size_t)d * HH + head] + 1e-9f);
  const float* hs = hp + (size_t)s * FDIM + lane * 8;
  float* od = out + (size_t)d * FDIM + lane * 8;
  float4 x0 = *(const float4*)hs, x1 = *(const float4*)(hs + 4);
  atomicAdd(od + 0, alpha * x0.x);
  atomicAdd(od + 1, alpha * x0.y);
  atomicAdd(od + 2, alpha * x0.z);
  atomicAdd(od + 3, alpha * x0.w);
  atomicAdd(od + 4, alpha * x1.x);
  atomicAdd(od + 5, alpha * x1.y);
  atomicAdd(od + 6, alpha * x1.z);
  atomicAdd(od + 7, alpha * x1.w);
}

__global__ void k_relu(float* __restrict__ p, int total) {
  int i = blockIdx.x * blockDim.x + threadIdx.x;
  if (i < total) p[i] = fmaxf(p[i], 0.f);
}

// ---------------- host-side orchestration ----------------
extern "C" void kernel_launch(void* const* d_in, const int* in_sizes, int n_in,
                              void* d_out, int out_size, void* d_ws, size_t ws_size,
                              hipStream_t stream) {
  (void)in_sizes; (void)n_in; (void)out_size; (void)ws_size;
  const float* x    = (const float*)d_in[0];
  const int*   src  = (const int*)d_in[1];
  const int*   dst  = (const int*)d_in[2];
  const float* W    = (const float*)d_in[3];
  const float* al   = (const float*)d_in[4];
  const float* ar   = (const float*)d_in[5];
  const float* bias = (const float*)d_in[6];
  const float* Wc   = (const float*)d_in[7];
  const float* bc   = (const float*)d_in[8];
  float* logits = (float*)d_out;

  char* ws = (char*)d_ws;
  const size_t fbytes = (size_t)NN * FDIM * sizeof(float);   // 30.72 MB
  float* hp  = (float*)(ws);
  float* h0  = (float*)(ws + fbytes);
  float* h1  = (float*)(ws + 2 * fbytes);
  float* el  = (float*)(ws + 3 * fbytes);
  float* er  = el + (size_t)NN * HH;
  int*   mi  = (int*)(er + (size_t)NN * HH);
  float* den = (float*)(mi + (size_t)NN * HH);
  float* eb  = den + (size_t)NN * HH;                        // [EE,HH]

  const float* hin = x;
  float* houts[2] = {h0, h1};

  for (int l = 0; l < LL; ++l) {
    float* hout = houts[l];
    k_init_out<<<(NN * FDIM + 255) / 256, 256, 0, stream>>>(
        hout, bias + (size_t)(l * RR + 0) * FDIM, bias + (size_t)(l * RR + 1) * FDIM,
        NN * FDIM);
    for (int r = 0; r < RR; ++r) {
      const size_t lr = (size_t)(l * RR + r);
      const float* Wlr = W + lr * FDIM * FDIM;
      dim3 gg((NN + 127) / 128, 4);
      k_gemm256<<<gg, 256, 0, stream>>>(hin, Wlr, hp, NN);
      k_attn_scores<<<(NN + 7) / 8, 256, 0, stream>>>(
          hp, al + lr * FDIM, ar + lr * FDIM, el, er, NN);
      k_init_nh<<<(NN * HH + 255) / 256, 256, 0, stream>>>(mi, den, NN * HH);
      const int* se = src + lr * EE;
      const int* de = dst + lr * EE;
      k_edge_max<<<(EE + 255) / 256, 256, 0, stream>>>(se, de, el, er, eb, mi, EE);
      k_edge_exp<<<(EE + 255) / 256, 256, 0, stream>>>(de, mi, eb, den, EE);
      k_edge_aggr<<<(EE + 7) / 8, 256, 0, stream>>>(se, de, eb, den, hp, hout, EE);
    }
    if (l == 0) k_relu<<<(NN * FDIM + 255) / 256, 256, 0, stream>>>(hout, NN * FDIM);
    hin = hout;
  }
  k_gemm_cls<<<(NN + 127) / 128, 256, 0, stream>>>(hin, Wc, bc, logits, NN);
}